// GPUMeshOptimizationOperator_68186900791880
// MI455X (gfx1250) — compile-verified
//
#include <hip/hip_runtime.h>
#include <hip/hip_bf16.h>
#include <stdint.h>

typedef float v4f __attribute__((ext_vector_type(4)));
typedef int   v4i __attribute__((ext_vector_type(4)));
typedef __attribute__((address_space(3))) int       lds_int;
typedef __attribute__((address_space(3))) v4i       lds_v4i;
typedef __attribute__((address_space(1))) v4i       gbl_v4i;

#define TPB 256
#define TILE_INTS (TPB * 3)   // 768 ints = 3 KB of indices per block tile

// ---------------------------------------------------------------------------
// gfx1250 feature probes (compile-time): async global->LDS copy + ASYNCcnt wait
// ---------------------------------------------------------------------------
#if defined(__HIP_DEVICE_COMPILE__) && defined(__gfx1250__)
#if __has_builtin(__builtin_amdgcn_global_load_async_to_lds_b128)
#define HAVE_ASYNC_B128 1
#endif
#endif

__device__ __forceinline__ void wait_async0() {
#if defined(__HIP_DEVICE_COMPILE__) && defined(__gfx1250__)
#if __has_builtin(__builtin_amdgcn_s_wait_asynccnt)
  __builtin_amdgcn_s_wait_asynccnt(0);
#else
  asm volatile("s_wait_asynccnt 0" ::: "memory");
#endif
#endif
}

// Device-scope f32 atomic add: executes at L2 (accumulator is L2-resident:
// 2M verts * 16B = 32MB << 192MB L2).
__device__ __forceinline__ void atomic_add_f32(float* p, float v) {
  __hip_atomic_fetch_add(p, v, __ATOMIC_RELAXED, __HIP_MEMORY_SCOPE_AGENT);
}

// ---------------------------------------------------------------------------
// HOT KERNEL (first in file so the disasm snippet shows it):
// per-triangle edge scatter.
//   Index tile staged block-wide into LDS via async b128 copies (ASYNCcnt
//   path, coalesced b128 traffic), read back via true ds_load (as3-typed
//   pointer, no generic/FLAT detour); then 3 random vertex gathers + 12 f32
//   atomics per triangle (all 4 atomics per target vertex in one cacheline).
// ---------------------------------------------------------------------------
__global__ __launch_bounds__(TPB) void scatter_kernel(const float* __restrict__ verts,
                                                      const int*   __restrict__ idx,
                                                      float*       __restrict__ acc,
                                                      int n_tris) {
  __shared__ int smem[TILE_INTS];
  const int tid = threadIdx.x;
  const long long tri0   = (long long)blockIdx.x * TPB;
  const long long base   = tri0 * 3;               // first index element of tile
  const long long n_ints = (long long)n_tris * 3;
  const long long t      = tri0 + tid;             // this thread's triangle

  int i0, i1, i2;
#ifdef HAVE_ASYNC_B128
  const bool full_tile = (base + TILE_INTS) <= n_ints;
  if (full_tile) {
    // LDS offset of smem (low 32 bits of its generic address == LDS offset)
    const uint32_t smem_off = (uint32_t)(uintptr_t)smem;
    if (tid < TILE_INTS / 4) {  // 192 lanes issue one b128 (16B) each
      __builtin_amdgcn_global_load_async_to_lds_b128(
          (gbl_v4i*)(uintptr_t)(idx + base + tid * 4),
          (lds_v4i*)(smem_off + tid * 16),
          /*offset=*/0, /*cpol=*/0);
    }
    wait_async0();      // this wave's async copies landed in LDS
    __syncthreads();    // all waves' copies landed
    // read back via addrspace(3)-typed pointer -> ds_load, not FLAT
    const lds_int* sp = (const lds_int*)smem_off;
    i0 = sp[3 * tid + 0];
    i1 = sp[3 * tid + 1];
    i2 = sp[3 * tid + 2];
  } else
#endif
  {
    if (t >= n_tris) return;
    // read-once stream: NT so it doesn't evict the L2-resident verts/acc
    i0 = __builtin_nontemporal_load(idx + 3 * t + 0);
    i1 = __builtin_nontemporal_load(idx + 3 * t + 1);
    i2 = __builtin_nontemporal_load(idx + 3 * t + 2);
  }
  if (t >= n_tris) return;

  // gather the three vertices (24MB vertex array is L2-resident after 1st touch)
  const float ax = verts[3LL * i0 + 0], ay = verts[3LL * i0 + 1], az = verts[3LL * i0 + 2];
  const float bx = verts[3LL * i1 + 0], by = verts[3LL * i1 + 1], bz = verts[3LL * i1 + 2];
  const float cx = verts[3LL * i2 + 0], cy = verts[3LL * i2 + 1], cz = verts[3LL * i2 + 2];

  // directed edges a->b, b->c, c->a : acc[v1] += vert[v2], count++
  float* p0 = acc + 4LL * i0;
  atomic_add_f32(p0 + 0, bx); atomic_add_f32(p0 + 1, by);
  atomic_add_f32(p0 + 2, bz); atomic_add_f32(p0 + 3, 1.0f);

  float* p1 = acc + 4LL * i1;
  atomic_add_f32(p1 + 0, cx); atomic_add_f32(p1 + 1, cy);
  atomic_add_f32(p1 + 2, cz); atomic_add_f32(p1 + 3, 1.0f);

  float* p2 = acc + 4LL * i2;
  atomic_add_f32(p2 + 0, ax); atomic_add_f32(p2 + 1, ay);
  atomic_add_f32(p2 + 2, az); atomic_add_f32(p2 + 3, 1.0f);
}

// ---------------------------------------------------------------------------
// Kernel 1: zero the accumulator (sum.xyz, count.w per vertex)
// ---------------------------------------------------------------------------
__global__ __launch_bounds__(TPB) void zero_acc_kernel(v4f* __restrict__ acc, int n_verts) {
  int i = blockIdx.x * TPB + threadIdx.x;
  if (i < n_verts) {
    v4f z = {0.0f, 0.0f, 0.0f, 0.0f};
    acc[i] = z;
  }
}

// ---------------------------------------------------------------------------
// Kernel 3: finalize smoothed = 0.5*v + 0.5*(count>0 ? sum/count : v)
// NT stores: write-once stream, keep accumulator/verts in L2.
// ---------------------------------------------------------------------------
__global__ __launch_bounds__(TPB) void finalize_kernel(const float* __restrict__ verts,
                                                       const v4f*   __restrict__ acc,
                                                       float*       __restrict__ out,
                                                       int n_verts) {
  int v = blockIdx.x * TPB + threadIdx.x;
  if (v >= n_verts) return;
  v4f a = acc[v];
  const float x = verts[3LL * v + 0];
  const float y = verts[3LL * v + 1];
  const float z = verts[3LL * v + 2];
  const float c = a.w;
  const float inv = 1.0f / fmaxf(c, 1.0f);
  const bool  has = c > 0.0f;
  const float mx = has ? a.x * inv : x;
  const float my = has ? a.y * inv : y;
  const float mz = has ? a.z * inv : z;
  __builtin_nontemporal_store(0.5f * (x + mx), &out[3LL * v + 0]);
  __builtin_nontemporal_store(0.5f * (y + my), &out[3LL * v + 1]);
  __builtin_nontemporal_store(0.5f * (z + mz), &out[3LL * v + 2]);
}

// ---------------------------------------------------------------------------
// Kernel 4: second tuple output = indices, promoted int32 -> float32
// (exact: all indices < 2^24). Vectorized NT b128 load + NT b128 store.
// ---------------------------------------------------------------------------
__global__ __launch_bounds__(TPB) void idxcast_kernel(const int* __restrict__ idx,
                                                      float*     __restrict__ out,
                                                      int n) {
  int i  = blockIdx.x * TPB + threadIdx.x;
  int i4 = i * 4;
  if (i4 + 3 < n) {
    v4i t = __builtin_nontemporal_load((const v4i*)(idx + i4));
    v4f f = {(float)t.x, (float)t.y, (float)t.z, (float)t.w};
    __builtin_nontemporal_store(f, (v4f*)(out + i4));
  } else {
    for (; i4 < n; ++i4)
      __builtin_nontemporal_store((float)__builtin_nontemporal_load(idx + i4), out + i4);
  }
}

// ---------------------------------------------------------------------------
extern "C" void kernel_launch(void* const* d_in, const int* in_sizes, int n_in,
                              void* d_out, int out_size, void* d_ws, size_t ws_size,
                              hipStream_t stream) {
  (void)n_in; (void)out_size; (void)ws_size;

  const float* verts = (const float*)d_in[0];
  const int*   idx   = (const int*)d_in[1];
  const int n_verts  = in_sizes[0] / 3;   // 2,000,000
  const int n_tris   = in_sizes[1] / 3;   // 4,000,000
  const int n_idx    = in_sizes[1];       // 12,000,000

  float* out = (float*)d_out;             // [smoothed (n_verts*3) | indices as f32 (n_idx)]
  float* acc = (float*)d_ws;              // n_verts * 4 floats (sum.xyz, count.w) = 32 MB

  const int gz = (n_verts + TPB - 1) / TPB;
  zero_acc_kernel<<<gz, TPB, 0, stream>>>((v4f*)acc, n_verts);

  const int gs = (n_tris + TPB - 1) / TPB;
  scatter_kernel<<<gs, TPB, 0, stream>>>(verts, idx, acc, n_tris);

  finalize_kernel<<<gz, TPB, 0, stream>>>(verts, (const v4f*)acc, out, n_verts);

  const int n4 = (n_idx + 3) / 4;
  const int gc = (n4 + TPB - 1) / TPB;
  idxcast_kernel<<<gc, TPB, 0, stream>>>(idx, out + (size_t)in_sizes[0], n_idx);
}